// SAM_4389456577505
// MI455X (gfx1250) — compile-verified
//
#include <hip/hip_runtime.h>
#include <math.h>

// MI455X / gfx1250, wave32. fp32 WMMA (16x16x4) for the main conv GEMM.
typedef __attribute__((ext_vector_type(2))) float v2f;
typedef __attribute__((ext_vector_type(8))) float v8f;

// Problem constants (from reference): B=16, C=64, H=W=256, G=8, gs=8, A=3.

// ---------------------------------------------------------------------------
// Kernel 1: one pass over x.
//   - s[b,g,h,w]   = sum of the 8 channels in group g       (float4 vectorized)
//   - partials[b,blk,c] = per-block spatial sum of channel c (shfl reduction)
// Grid (64, 16): 64 pixel-blocks of 1024 pixels per batch sample.
// ---------------------------------------------------------------------------
__global__ __launch_bounds__(256) void sam_reduce(const float* __restrict__ x,
                                                  float* __restrict__ s,
                                                  float* __restrict__ partials) {
  __shared__ float smem[8 * 64];  // [wave][channel]
  const int tid  = threadIdx.x;
  const int b    = blockIdx.y;
  const int blk  = blockIdx.x;
  const int lane = tid & 31;
  const int wave = tid >> 5;
  const long p = (long)blk * 1024 + (long)tid * 4;  // 4 consecutive pixels

  for (int g = 0; g < 8; ++g) {
    float4 gs = make_float4(0.f, 0.f, 0.f, 0.f);
#pragma unroll
    for (int cg = 0; cg < 8; ++cg) {
      const int c = g * 8 + cg;
      const float4 v = *(const float4*)(x + (((long)b * 64 + c) << 16) + p);
      gs.x += v.x; gs.y += v.y; gs.z += v.z; gs.w += v.w;
      // per-channel spatial partial: reduce 32 lanes (128 pixels)
      float cs = (v.x + v.y) + (v.z + v.w);
      cs += __shfl_xor(cs, 16, 32);
      cs += __shfl_xor(cs, 8, 32);
      cs += __shfl_xor(cs, 4, 32);
      cs += __shfl_xor(cs, 2, 32);
      cs += __shfl_xor(cs, 1, 32);
      if (lane == 0) smem[wave * 64 + c] = cs;
    }
    *(float4*)(s + (((long)b * 8 + g) << 16) + p) = gs;
  }
  __syncthreads();
  if (tid < 64) {
    float t = 0.f;
#pragma unroll
    for (int w = 0; w < 8; ++w) t += smem[w * 64 + tid];
    partials[(b * 64 + blk) * 64 + tid] = t;
  }
}

// ---------------------------------------------------------------------------
// Kernel 2 (single block): means -> Wk = tanh(means@w1^T+b1); w2g group-reduce.
//   wk [b][g*9+tap]        (72 per sample)
//   w2g[o][tap*8+g]        (K-ordering of the WMMA GEMM: K = tap*8 + g)
// ---------------------------------------------------------------------------
__global__ __launch_bounds__(256) void sam_small(const float* __restrict__ partials,
                                                 const float* __restrict__ w1,
                                                 const float* __restrict__ b1,
                                                 const float* __restrict__ w2,
                                                 float* __restrict__ wk,
                                                 float* __restrict__ w2g) {
  __shared__ float meansL[16 * 64];
  const int tid = threadIdx.x;
  for (int i = tid; i < 1024; i += 256) {
    const int b = i >> 6, c = i & 63;
    float t = 0.f;
    for (int blk = 0; blk < 64; ++blk) t += partials[(b * 64 + blk) * 64 + c];
    meansL[i] = t * (1.0f / 65536.0f);
  }
  __syncthreads();
  for (int i = tid; i < 16 * 72; i += 256) {  // Wk
    const int b = i / 72, k = i - b * 72;     // k = g*9 + tap (matches w1 rows)
    float t = b1[k];
    for (int c = 0; c < 64; ++c) t += meansL[b * 64 + c] * w1[k * 64 + c];
    wk[i] = tanhf(t);
  }
  for (int i = tid; i < 64 * 72; i += 256) {  // w2g, K = tap*8 + g
    const int o = i / 72, r = i - o * 72;
    const int tap = r >> 3, g = r & 7;
    float t = 0.f;
#pragma unroll
    for (int cg = 0; cg < 8; ++cg) t += w2[(o * 64 + g * 8 + cg) * 9 + tap];
    w2g[i] = t;
  }
}

// ---------------------------------------------------------------------------
// Kernel 3: fused depthwise-att + 3x3 conv as WMMA GEMM (M=pixels,N=64,K=72).
// Per block: 16x16 output tile for one b.
//   Phase 1: s tile (20x20x8, [y][x][g]) + wk + w2g -> LDS
//   Phase 2: att tile (18x18x8, [y][x][g]); att outside image = 0 (conv2 pad),
//            in-image att uses zero-padded s (conv1 pad).
//   Phase 3: each wave -> 2 pixel rows; 18 K-chunks x 4 N-tiles of
//            v_wmma_f32_16x16x4_f32; chunk kk covers K = tap*8 + gbase..+3.
// A-frag layout (ISA 16x4 f32): lanes 0-15 hold {K0,K1}, lanes 16-31 {K2,K3}.
// D layout: vgpr v, lane L: M = v + (L>=16)*8 (pixel x), N = L%16 (out ch).
// ---------------------------------------------------------------------------
__global__ __launch_bounds__(256) void sam_conv(const float* __restrict__ s,
                                                const float* __restrict__ wk,
                                                const float* __restrict__ w2g,
                                                const float* __restrict__ b2,
                                                float* __restrict__ out) {
  __shared__ __align__(16) float s_lds[20 * 20 * 8];
  __shared__ __align__(16) float att_lds[18 * 18 * 8];
  __shared__ __align__(16) float wk_lds[72];
  __shared__ __align__(16) float w2g_lds[64 * 72];

  const int tid = threadIdx.x;
  const int b   = blockIdx.z;
  const int y0  = blockIdx.y * 16;
  const int x0  = blockIdx.x * 16;

  // Phase 1: stage s tile (coalesced global reads), wk, w2g
  for (int i = tid; i < 3200; i += 256) {
    const int g = i / 400;
    const int rem = i - g * 400;
    const int sy = rem / 20, sx = rem - sy * 20;
    const int gy = y0 - 2 + sy, gx = x0 - 2 + sx;
    float v = 0.f;
    if (gy >= 0 && gy < 256 && gx >= 0 && gx < 256)
      v = s[(((long)b * 8 + g) << 16) + gy * 256 + gx];
    s_lds[(sy * 20 + sx) * 8 + g] = v;
  }
  if (tid < 72) wk_lds[tid] = wk[b * 72 + tid];
  for (int i = tid; i < 64 * 72; i += 256) w2g_lds[i] = w2g[i];
  __syncthreads();

  // Phase 2: att tile 18x18x8
  for (int i = tid; i < 2592; i += 256) {
    const int g  = i & 7;
    const int xy = i >> 3;
    const int ay = xy / 18, ax = xy - ay * 18;
    const int gy = y0 + ay - 1, gx = x0 + ax - 1;
    float v = 0.f;
    if (gy >= 0 && gy < 256 && gx >= 0 && gx < 256) {
#pragma unroll
      for (int ky = 0; ky < 3; ++ky)
#pragma unroll
        for (int kx = 0; kx < 3; ++kx)
          v += wk_lds[g * 9 + ky * 3 + kx] *
               s_lds[((ay + ky) * 20 + (ax + kx)) * 8 + g];
    }
    att_lds[i] = v;
  }
  __syncthreads();

  // Phase 3: WMMA. No divergence here (EXEC must be all-ones for WMMA).
  const int lane = tid & 31;
  const int wave = tid >> 5;
  const int m  = lane & 15;   // pixel (M) / out-channel (N) index within tile
  const int hi = lane >> 4;   // upper K-pair half

  for (int mt = 0; mt < 2; ++mt) {
    const int row = wave * 2 + mt;  // output row 0..15
    v8f acc[4] = {{}, {}, {}, {}};
#pragma unroll
    for (int kk = 0; kk < 18; ++kk) {
      const int tap = kk >> 1;
      const int dh = tap / 3 - 1;
      const int dw = (tap - (tap / 3) * 3) - 1;
      const int gh = (kk & 1) * 4 + hi * 2;  // group index of this K-pair
      const v2f a =
          *(const v2f*)&att_lds[((row + 1 + dh) * 18 + (m + 1 + dw)) * 8 + gh];
      const int k0 = kk * 4 + hi * 2;
#pragma unroll
      for (int nt = 0; nt < 4; ++nt) {
        const v2f bf = *(const v2f*)&w2g_lds[(nt * 16 + m) * 72 + k0];
        acc[nt] = __builtin_amdgcn_wmma_f32_16x16x4_f32(
            false, a, false, bf, (short)0, acc[nt], false, false);
      }
    }
    // Epilogue: +bias, two coalesced float4 stores per lane per N-tile
    const int gy  = y0 + row;
    const int gxb = x0 + hi * 8;
#pragma unroll
    for (int nt = 0; nt < 4; ++nt) {
      const int och = nt * 16 + m;
      const float bias = b2[och];
      float* op = out + (((long)b * 64 + och) << 16) + gy * 256 + gxb;
      float4 lo = make_float4(acc[nt][0] + bias, acc[nt][1] + bias,
                              acc[nt][2] + bias, acc[nt][3] + bias);
      float4 hi4 = make_float4(acc[nt][4] + bias, acc[nt][5] + bias,
                               acc[nt][6] + bias, acc[nt][7] + bias);
      *(float4*)op = lo;
      *(float4*)(op + 4) = hi4;
    }
  }
}

// ---------------------------------------------------------------------------
extern "C" void kernel_launch(void* const* d_in, const int* in_sizes, int n_in,
                              void* d_out, int out_size, void* d_ws, size_t ws_size,
                              hipStream_t stream) {
  const float* x  = (const float*)d_in[0];  // [16,64,256,256]
  const float* w1 = (const float*)d_in[1];  // [72,64,1,1]
  const float* b1 = (const float*)d_in[2];  // [72]
  const float* w2 = (const float*)d_in[3];  // [64,64,3,3]
  const float* b2 = (const float*)d_in[4];  // [64]
  float* out = (float*)d_out;               // [16,64,256,256]

  char* ws = (char*)d_ws;
  float* sbuf     = (float*)(ws);                                 // 32 MiB
  float* partials = (float*)(ws + 33554432);                      // 256 KiB
  float* wkbuf    = (float*)(ws + 33554432 + 262144);             // 1152 f
  float* w2gbuf   = (float*)(ws + 33554432 + 262144 + 8192);      // 4608 f

  sam_reduce<<<dim3(64, 16), 256, 0, stream>>>(x, sbuf, partials);
  sam_small<<<dim3(1), 256, 0, stream>>>(partials, w1, b1, w2, wkbuf, w2gbuf);
  sam_conv<<<dim3(16, 16, 16), 256, 0, stream>>>(sbuf, wkbuf, w2gbuf, b2, out);
}